// GAT_12567074308927
// MI455X (gfx1250) — compile-verified
//
#include <hip/hip_runtime.h>
#include <hip/hip_bf16.h>
#include <math.h>

#define NEG_ATT 0.2f
#define NEG_OUT 0.01f

typedef __attribute__((ext_vector_type(2))) float v2f;
typedef __attribute__((ext_vector_type(8))) float v8f;

// Sign-aware float atomic max (correct for mixed signs given -inf init).
__device__ __forceinline__ void atomic_max_float(float* addr, float val) {
    if (val >= 0.0f) {
        atomicMax((int*)addr, __float_as_int(val));
    } else {
        atomicMin((unsigned int*)addr, __float_as_uint(val));
    }
}

// ---------------------------------------------------------------------------
// 0) init: out = 0, seg_max = -inf, seg_sum = 0
// ---------------------------------------------------------------------------
__global__ void gat_init(float* __restrict__ out, float* __restrict__ seg_max,
                         float* __restrict__ seg_sum, int N) {
    int i = blockIdx.x * blockDim.x + threadIdx.x;
    int total = N * 32;
    if (i < total) out[i] = 0.0f;
    if (i < N) {
        ((unsigned int*)seg_max)[i] = 0xFF800000u; // -inf bits
        seg_sum[i] = 0.0f;
    }
}

// ---------------------------------------------------------------------------
// 1) Fused node GEMM: xl = x@Wl + bl, xr = x@Wr + br using fp32 WMMA 16x16x4.
//    One wave per 16-row node tile; 2 N-tiles (cols 0-15, 16-31) per matrix.
//    A frag (16x4 f32): lane m=lane&15; VGPR pair holds K = k0+2*(lane>>4)
//    and K+1.  B frag (4x16): lane n=lane&15, same K split.
//    C/D (16x16 f32): VGPR v holds M = v + 8*(lane>>4), N = lane&15.
// ---------------------------------------------------------------------------
__global__ __launch_bounds__(256) void gat_gemm(
        const float* __restrict__ x,
        const float* __restrict__ Wl, const float* __restrict__ bl,
        const float* __restrict__ Wr, const float* __restrict__ br,
        float* __restrict__ xl, float* __restrict__ xr, int N) {
    const int lane = threadIdx.x & 31;
    const int wave = threadIdx.x >> 5;
    const int tile = blockIdx.x * (blockDim.x >> 5) + wave;
    const int row0 = tile * 16;
    if (row0 >= N) return;

    const int m  = lane & 15;
    const int kh = lane >> 4;           // 0/1 -> K offset 0/2

    int arow = row0 + m;
    if (arow >= N) arow = N - 1;        // tail clamp (loads only)
    const float* __restrict__ xrow = x + (size_t)arow * 128;

    v8f cl0 = {}, cl1 = {}, cr0 = {}, cr1 = {};

    #pragma unroll 4
    for (int k0 = 0; k0 < 128; k0 += 4) {
        const int kb = k0 + 2 * kh;
        v2f a;
        a[0] = xrow[kb];
        a[1] = xrow[kb + 1];

        const int n = m;
        v2f bl0v, bl1v, br0v, br1v;
        bl0v[0] = Wl[kb * 32 + n];       bl0v[1] = Wl[(kb + 1) * 32 + n];
        bl1v[0] = Wl[kb * 32 + n + 16];  bl1v[1] = Wl[(kb + 1) * 32 + n + 16];
        br0v[0] = Wr[kb * 32 + n];       br0v[1] = Wr[(kb + 1) * 32 + n];
        br1v[0] = Wr[kb * 32 + n + 16];  br1v[1] = Wr[(kb + 1) * 32 + n + 16];

        cl0 = __builtin_amdgcn_wmma_f32_16x16x4_f32(false, a, false, bl0v, (short)0, cl0, false, false);
        cl1 = __builtin_amdgcn_wmma_f32_16x16x4_f32(false, a, false, bl1v, (short)0, cl1, false, false);
        cr0 = __builtin_amdgcn_wmma_f32_16x16x4_f32(false, a, false, br0v, (short)0, cr0, false, false);
        cr1 = __builtin_amdgcn_wmma_f32_16x16x4_f32(false, a, false, br1v, (short)0, cr1, false, false);
    }

    const int n = lane & 15;
    const float bl_lo = bl[n], bl_hi = bl[n + 16];
    const float br_lo = br[n], br_hi = br[n + 16];
    #pragma unroll
    for (int v = 0; v < 8; ++v) {
        const int mm = v + 8 * kh;
        const int row = row0 + mm;
        if (row < N) {
            const size_t ro = (size_t)row * 32;
            xl[ro + n]      = cl0[v] + bl_lo;
            xl[ro + n + 16] = cl1[v] + bl_hi;
            xr[ro + n]      = cr0[v] + br_lo;
            xr[ro + n + 16] = cr1[v] + br_hi;
        }
    }
}

// ---------------------------------------------------------------------------
// 2) Per-edge attention logits + segment max (L2-resident gathers).
// ---------------------------------------------------------------------------
__global__ __launch_bounds__(256) void gat_edge_logits(
        const int* __restrict__ src, const int* __restrict__ dst,
        const float* __restrict__ ew,
        const float* __restrict__ xl, const float* __restrict__ xr,
        const float* __restrict__ We, const float* __restrict__ att,
        float* __restrict__ logits, float* __restrict__ seg_max, int E) {
    int e = blockIdx.x * blockDim.x + threadIdx.x;
    if (e >= E) return;
    const int s = src[e], d = dst[e];
    const float w = ew[e];
    const float4* __restrict__ pl = (const float4*)(xl + (size_t)s * 32);
    const float4* __restrict__ pr = (const float4*)(xr + (size_t)d * 32);
    const float4* __restrict__ pe = (const float4*)We;
    const float4* __restrict__ pa = (const float4*)att;
    float acc = 0.0f;
    #pragma unroll
    for (int i = 0; i < 8; ++i) {
        float4 a = pl[i], b = pr[i], c = pe[i], t = pa[i];
        float m0 = a.x + b.x + w * c.x; m0 = (m0 > 0.0f) ? m0 : NEG_ATT * m0;
        float m1 = a.y + b.y + w * c.y; m1 = (m1 > 0.0f) ? m1 : NEG_ATT * m1;
        float m2 = a.z + b.z + w * c.z; m2 = (m2 > 0.0f) ? m2 : NEG_ATT * m2;
        float m3 = a.w + b.w + w * c.w; m3 = (m3 > 0.0f) ? m3 : NEG_ATT * m3;
        acc += m0 * t.x + m1 * t.y + m2 * t.z + m3 * t.w;
    }
    logits[e] = acc;
    atomic_max_float(&seg_max[d], acc);
}

// ---------------------------------------------------------------------------
// 3) ex = exp(logit - seg_max[dst]); overwrite logits with ex; seg_sum += ex
// ---------------------------------------------------------------------------
__global__ __launch_bounds__(256) void gat_edge_exp(
        const int* __restrict__ dst, float* __restrict__ logits,
        const float* __restrict__ seg_max, float* __restrict__ seg_sum, int E) {
    int e = blockIdx.x * blockDim.x + threadIdx.x;
    if (e >= E) return;
    const int d = dst[e];
    const float ex = __expf(logits[e] - seg_max[d]);
    logits[e] = ex;
    atomicAdd(&seg_sum[d], ex);
}

// ---------------------------------------------------------------------------
// 4) out[dst] += alpha * xl[src]   (alpha = ex / seg_sum[dst])
// ---------------------------------------------------------------------------
__global__ __launch_bounds__(256) void gat_edge_scatter(
        const int* __restrict__ src, const int* __restrict__ dst,
        const float* __restrict__ ex, const float* __restrict__ seg_sum,
        const float* __restrict__ xl, float* __restrict__ out, int E) {
    int e = blockIdx.x * blockDim.x + threadIdx.x;
    if (e >= E) return;
    const int s = src[e], d = dst[e];
    const float alpha = ex[e] / seg_sum[d];
    const float4* __restrict__ pl = (const float4*)(xl + (size_t)s * 32);
    float* __restrict__ po = out + (size_t)d * 32;
    #pragma unroll
    for (int i = 0; i < 8; ++i) {
        float4 v = pl[i];
        atomicAdd(po + 4 * i + 0, v.x * alpha);
        atomicAdd(po + 4 * i + 1, v.y * alpha);
        atomicAdd(po + 4 * i + 2, v.z * alpha);
        atomicAdd(po + 4 * i + 3, v.w * alpha);
    }
}

// ---------------------------------------------------------------------------
// 5) out = leaky_relu(out + bias, 0.01)
// ---------------------------------------------------------------------------
__global__ void gat_finalize(float* __restrict__ out,
                             const float* __restrict__ bias, int total) {
    int i = blockIdx.x * blockDim.x + threadIdx.x;
    if (i >= total) return;
    float v = out[i] + bias[i & 31];
    out[i] = (v > 0.0f) ? v : NEG_OUT * v;
}

extern "C" void kernel_launch(void* const* d_in, const int* in_sizes, int n_in,
                              void* d_out, int out_size, void* d_ws, size_t ws_size,
                              hipStream_t stream) {
    const float* x    = (const float*)d_in[0];
    const int*   ei   = (const int*)d_in[1];
    const float* ew   = (const float*)d_in[2];
    const float* Wl   = (const float*)d_in[3];
    const float* bl   = (const float*)d_in[4];
    const float* Wr   = (const float*)d_in[5];
    const float* br   = (const float*)d_in[6];
    const float* We   = (const float*)d_in[7];
    const float* att  = (const float*)d_in[8];
    const float* bias = (const float*)d_in[9];

    const int N = in_sizes[0] / 128;
    const int E = in_sizes[2];
    const int* srcv = ei;       // edge_index[0]
    const int* dstv = ei + E;   // edge_index[1]

    // workspace layout (floats): xl[N*32] | xr[N*32] | logits/ex[E] | seg_max[N] | seg_sum[N]
    float* xl      = (float*)d_ws;
    float* xr      = xl + (size_t)N * 32;
    float* logits  = xr + (size_t)N * 32;
    float* seg_max = logits + (size_t)E;
    float* seg_sum = seg_max + (size_t)N;
    float* out     = (float*)d_out;

    const int tpb = 256;

    gat_init<<<(N * 32 + tpb - 1) / tpb, tpb, 0, stream>>>(out, seg_max, seg_sum, N);

    const int tiles = (N + 15) / 16;
    const int wavesPerBlock = 8;
    gat_gemm<<<(tiles + wavesPerBlock - 1) / wavesPerBlock, 32 * wavesPerBlock, 0, stream>>>(
        x, Wl, bl, Wr, br, xl, xr, N);

    gat_edge_logits<<<(E + tpb - 1) / tpb, tpb, 0, stream>>>(
        srcv, dstv, ew, xl, xr, We, att, logits, seg_max, E);

    gat_edge_exp<<<(E + tpb - 1) / tpb, tpb, 0, stream>>>(
        dstv, logits, seg_max, seg_sum, E);

    gat_edge_scatter<<<(E + tpb - 1) / tpb, tpb, 0, stream>>>(
        srcv, dstv, logits, seg_sum, xl, out, E);

    gat_finalize<<<(N * 32 + tpb - 1) / tpb, tpb, 0, stream>>>(out, bias, N * 32);
}